// STGCN3_30897994728052
// MI455X (gfx1250) — compile-verified
//
#include <hip/hip_runtime.h>

typedef __attribute__((ext_vector_type(2))) float v2f;
typedef __attribute__((ext_vector_type(8))) float v8f;

#define BN_EPS 1e-5f

// ---------------------------------------------------------------------------
// Activations are channels-last [N, T, V, Cpad] fp32, Cpad = C rounded to 4.
// Weights pre-packed per stage into [Cout_pad64][K] row-major (k = tap*Cin+c)
// so each lane's B fragment is a contiguous 8-byte load.
// WMMA kernels: 256 threads = 8 waves (4 M x 2 N), each wave owns a 2x2 grid
// of 16x16 tiles (32M x 32N) -> one A pair + one B pair feed 4 wmma ops.
// ---------------------------------------------------------------------------

__global__ void k_zero(float* __restrict__ p, int n) {
    int i = blockIdx.x * blockDim.x + threadIdx.x;
    if (i < n) p[i] = 0.f;
}

// NCHW (N,C,T,V) -> channels-last (N,T,V,Cp), zero-filling padded channels
__global__ void k_in_pad(const float* __restrict__ in, float* __restrict__ out,
                         int N, int C, int Cp, int T, int V) {
    int idx = blockIdx.x * blockDim.x + threadIdx.x;
    int total = N * T * V * Cp;
    if (idx >= total) return;
    int c = idx % Cp;
    int v = (idx / Cp) % V;
    int t = (idx / (Cp * V)) % T;
    int n = idx / (Cp * V * T);
    out[idx] = (c < C) ? in[((long)(n * C + c) * T + t) * V + v] : 0.f;
}

// OIHW (CoutT,CinT,3,3) -> [CoutP][9*CinP], k = (dt*3+dv)*CinP + c, zero pad
__global__ void k_pack_w3x3(const float* __restrict__ w, float* __restrict__ wp,
                            int CoutT, int CoutP, int CinT, int CinP) {
    int idx = blockIdx.x * blockDim.x + threadIdx.x;
    int total = CoutP * 9 * CinP;
    if (idx >= total) return;
    int c = idx % CinP;
    int tap = (idx / CinP) % 9;
    int n = idx / (CinP * 9);
    wp[idx] = (n < CoutT && c < CinT) ? w[((long)(n * CinT + c)) * 9 + tap] : 0.f;
}

// (CoutT,CinT) -> [CoutP][CinP], zero pad
__global__ void k_pack_w1x1(const float* __restrict__ w, float* __restrict__ wp,
                            int CoutT, int CoutP, int CinT, int CinP) {
    int idx = blockIdx.x * blockDim.x + threadIdx.x;
    int total = CoutP * CinP;
    if (idx >= total) return;
    int c = idx % CinP;
    int n = idx / CinP;
    wp[idx] = (n < CoutT && c < CinT) ? w[(long)n * CinT + c] : 0.f;
}

// OIHW (CoutT,CinT,kt,1) -> [CoutP][kt*CinP], k = dt*CinP + c, zero pad
__global__ void k_pack_wdown(const float* __restrict__ w, float* __restrict__ wp,
                             int CoutT, int CoutP, int CinT, int CinP, int kt) {
    int idx = blockIdx.x * blockDim.x + threadIdx.x;
    int total = CoutP * kt * CinP;
    if (idx >= total) return;
    int c = idx % CinP;
    int dt = (idx / CinP) % kt;
    int n = idx / (CinP * kt);
    wp[idx] = (n < CoutT && c < CinT) ? w[((long)(n * CinT + c)) * kt + dt] : 0.f;
}

// h1[n,q,v,c] = sum_t x[n,t,v,c] * Tm[v,t,q]   (small einsum: VALU)
__global__ void k_gcn_t(const float* __restrict__ x, const float* __restrict__ Tm,
                        float* __restrict__ h, int N, int T, int V, int C) {
    int idx = blockIdx.x * blockDim.x + threadIdx.x;
    int total = N * T * V * C;
    if (idx >= total) return;
    int c = idx % C;
    int v = (idx / C) % V;
    int q = (idx / (C * V)) % T;
    int n = idx / (C * V * T);
    const float* xp = x + ((long)(n * T) * V + v) * C + c;   // t-stride V*C
    const float* tp = Tm + (long)(v * T) * T + q;            // t-stride T
    float acc = 0.f;
    for (int t = 0; t < T; ++t)
        acc = fmaf(xp[(long)t * V * C], tp[(long)t * T], acc);
    h[idx] = acc;
}

// h2[n,t,w,c] = sum_v h1[n,t,v,c] * A[t,v,w]   (small einsum: VALU)
__global__ void k_gcn_a(const float* __restrict__ h, const float* __restrict__ Am,
                        float* __restrict__ o, int N, int T, int V, int C) {
    int idx = blockIdx.x * blockDim.x + threadIdx.x;
    int total = N * T * V * C;
    if (idx >= total) return;
    int c = idx % C;
    int w = (idx / C) % V;
    int t = (idx / (C * V)) % T;
    int n = idx / (C * V * T);
    const float* hp = h + ((long)(n * T + t) * V) * C + c;   // v-stride C
    const float* ap = Am + (long)(t * V) * V + w;            // v-stride V
    float acc = 0.f;
    for (int v = 0; v < V; ++v)
        acc = fmaf(hp[(long)v * C], ap[(long)v * V], acc);
    o[idx] = acc;
}

#define WMMA_F32(A, B, C) \
    __builtin_amdgcn_wmma_f32_16x16x4_f32(false, (A), false, (B), (short)0, (C), false, false)

// ---------------------------------------------------------------------------
// 3x3 conv over (T,V), pad 1 both dims, + bias + BN + residual + PReLU
// ---------------------------------------------------------------------------
__global__ void __launch_bounds__(256)
k_conv3x3_wmma(const float* __restrict__ in,      // [N,T,V,Cin]
               const float* __restrict__ wp,      // packed [CoutP64][9*Cin]
               const float* __restrict__ bias,
               const float* __restrict__ g, const float* __restrict__ be,
               const float* __restrict__ mu, const float* __restrict__ va,
               const float* __restrict__ res,     // [M][CoutS]
               const float* __restrict__ prelu,
               const float* __restrict__ zpage,   // >= Cin+4 zeros
               float* __restrict__ out,           // [M][CoutS]
               int N, int T, int V, int Cin, int CoutT, int CoutS) {
    const int lane = threadIdx.x & 31;
    const int wave = threadIdx.x >> 5;
    const int mb = blockIdx.x * 128 + (wave >> 1) * 32;
    const int nb = blockIdx.y * 64 + (wave & 1) * 32;
    const int lr = lane & 15;
    const int lh = lane >> 4;
    const int co = 2 * lh;                 // lane's K sub-offset

    const int m0 = mb + lr, m1 = m0 + 16;
    const int vA0 = m0 % V, tA0 = (m0 / V) % T, nA0 = m0 / (V * T);
    const int vA1 = m1 % V, tA1 = (m1 / V) % T, nA1 = m1 / (V * T);

    const int nn = nb + lr;
    const int K = Cin * 9;
    const float* wb0 = wp + (long)nn * K + co;
    const float* wb1 = wb0 + (long)16 * K;

    const long rowst = (long)V * Cin;      // t-stride
    const float* inb0 = in + (long)nA0 * T * rowst;
    const float* inb1 = in + (long)nA1 * T * rowst;

    v8f acc[2][2] = {};
    int kbase = 0;
    for (int dt = 0; dt < 3; ++dt) {
        const int ti0 = tA0 + dt - 1, ti1 = tA1 + dt - 1;
        const bool tok0 = ((unsigned)ti0 < (unsigned)T);
        const bool tok1 = ((unsigned)ti1 < (unsigned)T);
        for (int dv = 0; dv < 3; ++dv) {
            const int vi0 = vA0 + dv - 1, vi1 = vA1 + dv - 1;
            const bool ok0 = tok0 && ((unsigned)vi0 < (unsigned)V);
            const bool ok1 = tok1 && ((unsigned)vi1 < (unsigned)V);
            const float* ap0 = ok0 ? (inb0 + (long)ti0 * rowst + (long)vi0 * Cin + co)
                                   : (zpage + co);
            const float* ap1 = ok1 ? (inb1 + (long)ti1 * rowst + (long)vi1 * Cin + co)
                                   : (zpage + co);
            const float* bp0 = wb0 + kbase;
            const float* bp1 = wb1 + kbase;
            for (int c = 0; c < Cin; c += 4) {
                v2f a0 = *(const v2f*)(ap0 + c);
                v2f a1 = *(const v2f*)(ap1 + c);
                v2f b0 = *(const v2f*)(bp0 + c);
                v2f b1 = *(const v2f*)(bp1 + c);
                acc[0][0] = WMMA_F32(a0, b0, acc[0][0]);
                acc[0][1] = WMMA_F32(a0, b1, acc[0][1]);
                acc[1][0] = WMMA_F32(a1, b0, acc[1][0]);
                acc[1][1] = WMMA_F32(a1, b1, acc[1][1]);
            }
            kbase += Cin;
        }
    }

    const float al = prelu[0];
#pragma unroll
    for (int j = 0; j < 2; ++j) {
        const int no = nb + 16 * j + lr;
        if (no < CoutT) {
            const float sc = g[no] * rsqrtf(va[no] + BN_EPS);
            const float sh = be[no] - mu[no] * sc;
            const float bi = bias[no];
#pragma unroll
            for (int i = 0; i < 2; ++i) {
                const int mrow = mb + 16 * i + 8 * lh;
#pragma unroll
                for (int r = 0; r < 8; ++r) {
                    const long mm = mrow + r;
                    float y = (acc[i][j][r] + bi) * sc + sh + res[mm * CoutS + no];
                    out[mm * CoutS + no] = (y >= 0.f) ? y : al * y;
                }
            }
        } else if (no < CoutS) {           // keep padded channels zeroed
#pragma unroll
            for (int i = 0; i < 2; ++i) {
                const int mrow = mb + 16 * i + 8 * lh;
#pragma unroll
                for (int r = 0; r < 8; ++r)
                    out[(long)(mrow + r) * CoutS + no] = 0.f;
            }
        }
    }
}

// ---------------------------------------------------------------------------
// 1x1 conv (pure GEMM) + bias + BN  (residual branch)
// ---------------------------------------------------------------------------
__global__ void __launch_bounds__(256)
k_conv1x1_wmma(const float* __restrict__ in,      // [M][Cin]
               const float* __restrict__ wp,      // packed [CoutP64][Cin]
               const float* __restrict__ bias,
               const float* __restrict__ g, const float* __restrict__ be,
               const float* __restrict__ mu, const float* __restrict__ va,
               float* __restrict__ out,           // [M][CoutS]
               int M, int Cin, int CoutT, int CoutS) {
    const int lane = threadIdx.x & 31;
    const int wave = threadIdx.x >> 5;
    const int mb = blockIdx.x * 128 + (wave >> 1) * 32;
    const int nb = blockIdx.y * 64 + (wave & 1) * 32;
    const int lr = lane & 15;
    const int lh = lane >> 4;
    const int co = 2 * lh;

    const float* ap0 = in + (long)(mb + lr) * Cin + co;
    const float* ap1 = ap0 + (long)16 * Cin;
    const float* bp0 = wp + (long)(nb + lr) * Cin + co;
    const float* bp1 = bp0 + (long)16 * Cin;

    v8f acc[2][2] = {};
    for (int c = 0; c < Cin; c += 4) {
        v2f a0 = *(const v2f*)(ap0 + c);
        v2f a1 = *(const v2f*)(ap1 + c);
        v2f b0 = *(const v2f*)(bp0 + c);
        v2f b1 = *(const v2f*)(bp1 + c);
        acc[0][0] = WMMA_F32(a0, b0, acc[0][0]);
        acc[0][1] = WMMA_F32(a0, b1, acc[0][1]);
        acc[1][0] = WMMA_F32(a1, b0, acc[1][0]);
        acc[1][1] = WMMA_F32(a1, b1, acc[1][1]);
    }

#pragma unroll
    for (int j = 0; j < 2; ++j) {
        const int no = nb + 16 * j + lr;
        if (no < CoutT) {
            const float sc = g[no] * rsqrtf(va[no] + BN_EPS);
            const float sh = be[no] - mu[no] * sc;
            const float bi = bias[no];
#pragma unroll
            for (int i = 0; i < 2; ++i) {
                const int mrow = mb + 16 * i + 8 * lh;
#pragma unroll
                for (int r = 0; r < 8; ++r)
                    out[(long)(mrow + r) * CoutS + no] = (acc[i][j][r] + bi) * sc + sh;
            }
        } else if (no < CoutS) {
#pragma unroll
            for (int i = 0; i < 2; ++i) {
                const int mrow = mb + 16 * i + 8 * lh;
#pragma unroll
                for (int r = 0; r < 8; ++r)
                    out[(long)(mrow + r) * CoutS + no] = 0.f;
            }
        }
    }
}

// ---------------------------------------------------------------------------
// temporal conv (kt,1), stride (stride,1), pad (pad,0), no bias, +BN +ReLU
// ---------------------------------------------------------------------------
__global__ void __launch_bounds__(256)
k_down_wmma(const float* __restrict__ in,         // [N,Tin,V,Cin]
            const float* __restrict__ wp,         // packed [CoutP64][kt*Cin]
            const float* __restrict__ g, const float* __restrict__ be,
            const float* __restrict__ mu, const float* __restrict__ va,
            const float* __restrict__ zpage,
            float* __restrict__ out,              // [M][CoutS]
            int N, int Tin, int Tout, int V, int Cin,
            int CoutT, int CoutS, int kt, int stride, int pad) {
    const int lane = threadIdx.x & 31;
    const int wave = threadIdx.x >> 5;
    const int mb = blockIdx.x * 128 + (wave >> 1) * 32;
    const int nb = blockIdx.y * 64 + (wave & 1) * 32;
    const int lr = lane & 15;
    const int lh = lane >> 4;
    const int co = 2 * lh;

    const int m0 = mb + lr, m1 = m0 + 16;
    const int vA0 = m0 % V, to0 = (m0 / V) % Tout, nA0 = m0 / (V * Tout);
    const int vA1 = m1 % V, to1 = (m1 / V) % Tout, nA1 = m1 / (V * Tout);

    const int nn = nb + lr;
    const int K = Cin * kt;
    const float* wb0 = wp + (long)nn * K + co;
    const float* wb1 = wb0 + (long)16 * K;

    const long rowst = (long)V * Cin;
    const float* inb0 = in + (long)nA0 * Tin * rowst + (long)vA0 * Cin + co;
    const float* inb1 = in + (long)nA1 * Tin * rowst + (long)vA1 * Cin + co;

    v8f acc[2][2] = {};
    for (int dt = 0; dt < kt; ++dt) {
        const int ti0 = to0 * stride + dt - pad;
        const int ti1 = to1 * stride + dt - pad;
        const bool ok0 = ((unsigned)ti0 < (unsigned)Tin);
        const bool ok1 = ((unsigned)ti1 < (unsigned)Tin);
        const float* ap0 = ok0 ? (inb0 + (long)ti0 * rowst) : (zpage + co);
        const float* ap1 = ok1 ? (inb1 + (long)ti1 * rowst) : (zpage + co);
        const float* bp0 = wb0 + dt * Cin;
        const float* bp1 = wb1 + dt * Cin;
        for (int c = 0; c < Cin; c += 4) {
            v2f a0 = *(const v2f*)(ap0 + c);
            v2f a1 = *(const v2f*)(ap1 + c);
            v2f b0 = *(const v2f*)(bp0 + c);
            v2f b1 = *(const v2f*)(bp1 + c);
            acc[0][0] = WMMA_F32(a0, b0, acc[0][0]);
            acc[0][1] = WMMA_F32(a0, b1, acc[0][1]);
            acc[1][0] = WMMA_F32(a1, b0, acc[1][0]);
            acc[1][1] = WMMA_F32(a1, b1, acc[1][1]);
        }
    }

#pragma unroll
    for (int j = 0; j < 2; ++j) {
        const int no = nb + 16 * j + lr;
        if (no < CoutT) {
            const float sc = g[no] * rsqrtf(va[no] + BN_EPS);
            const float sh = be[no] - mu[no] * sc;
#pragma unroll
            for (int i = 0; i < 2; ++i) {
                const int mrow = mb + 16 * i + 8 * lh;
#pragma unroll
                for (int r = 0; r < 8; ++r) {
                    float y = acc[i][j][r] * sc + sh;
                    out[(long)(mrow + r) * CoutS + no] = (y > 0.f) ? y : 0.f;
                }
            }
        } else if (no < CoutS) {
#pragma unroll
            for (int i = 0; i < 2; ++i) {
                const int mrow = mb + 16 * i + 8 * lh;
#pragma unroll
                for (int r = 0; r < 8; ++r)
                    out[(long)(mrow + r) * CoutS + no] = 0.f;
            }
        }
    }
}

// ---------------------------------------------------------------------------
// Host orchestration
// ---------------------------------------------------------------------------
extern "C" void kernel_launch(void* const* d_in, const int* in_sizes, int n_in,
                              void* d_out, int out_size, void* d_ws, size_t ws_size,
                              hipStream_t stream) {
    (void)in_sizes; (void)n_in; (void)out_size; (void)ws_size;

    const int N = 256, V = 14;
    const int DIMS[4][3] = {{2,128,64},{128,128,32},{128,256,16},{256,2,8}};

    auto f = [&](int i) { return (const float*)d_in[i]; };
    auto pad4  = [](int c) { return (c + 3) & ~3; };
    auto pad64 = [](int c) { return (c + 63) & ~63; };

    const size_t MB = 1024ull * 1024ull;
    char* ws = (char*)d_ws;
    float* X   = (float*)(ws + 0);          //  64 MB: current activation
    float* H1  = (float*)(ws + 64  * MB);   //  64 MB: gcn temporal-mix
    float* H2  = (float*)(ws + 128 * MB);   //  64 MB: gcn joint-mix
    float* Y   = (float*)(ws + 192 * MB);   // 128 MB: stage output
    float* R   = (float*)(ws + 320 * MB);   // 128 MB: residual branch
    float* WP3 = (float*)(ws + 448 * MB);   //   4 MB: packed 3x3 weights
    float* WP1 = (float*)(ws + 452 * MB);   //   1 MB: packed 1x1 weights
    float* WPD = (float*)(ws + 453 * MB);   //   4 MB: packed down weights
    float* ZP  = (float*)(ws + 457 * MB);   //   4 KB: zero page

    // Parse d_in per setup_inputs() dict insertion order.
    struct SP {
        const float *Tm, *Am, *cw, *cb, *g, *b, *m, *v;
        const float *rw, *rb, *rg, *rbb, *rm, *rv;
        const float *pr, *dw, *dg, *db, *dm, *dv;
        bool hr;
    } sp[4];
    int ci = 1;  // d_in[0] = x
    for (int s = 0; s < 4; ++s) {
        SP& p = sp[s];
        p.Tm = f(ci++); p.Am = f(ci++);
        p.cw = f(ci++); p.cb = f(ci++);
        p.g = f(ci++); p.b = f(ci++); p.m = f(ci++); p.v = f(ci++);
        p.hr = (DIMS[s][0] != DIMS[s][1]);
        if (p.hr) {
            p.rw = f(ci++); p.rb = f(ci++);
            p.rg = f(ci++); p.rbb = f(ci++); p.rm = f(ci++); p.rv = f(ci++);
        }
        p.pr = f(ci++); p.dw = f(ci++);
        p.dg = f(ci++); p.db = f(ci++); p.dm = f(ci++); p.dv = f(ci++);
    }

    k_zero<<<2, 256, 0, stream>>>(ZP, 512);

    // Input (N,2,64,V) NCHW -> channels-last padded to Cp=4
    {
        int total = N * 64 * V * 4;
        k_in_pad<<<(total + 255) / 256, 256, 0, stream>>>(
            (const float*)d_in[0], X, N, 2, 4, 64, V);
    }

    int T = 64;
    for (int s = 0; s < 4; ++s) {
        const int cinT = DIMS[s][0], coutT = DIMS[s][1];
        const int cinP = pad4(cinT);          // activation channel stride (in)
        const int coutS = pad4(coutT);        // activation channel stride (out)
        const int coutP = pad64(coutT);       // packed-weight row coverage
        SP& p = sp[s];

        // gcn: temporal mix then joint mix (padded channels stay zero)
        {
            int total = N * T * V * cinP;
            k_gcn_t<<<(total + 255) / 256, 256, 0, stream>>>(X, p.Tm, H1, N, T, V, cinP);
            k_gcn_a<<<(total + 255) / 256, 256, 0, stream>>>(H1, p.Am, H2, N, T, V, cinP);
        }

        const int M = N * T * V;              // always a multiple of 128

        // residual branch
        const float* res = X;                 // identity (cin == cout)
        if (p.hr) {
            int tot = coutP * cinP;
            k_pack_w1x1<<<(tot + 255) / 256, 256, 0, stream>>>(
                p.rw, WP1, coutT, coutP, cinT, cinP);
            dim3 g1(M / 128, coutP / 64);
            k_conv1x1_wmma<<<g1, 256, 0, stream>>>(X, WP1, p.rb,
                                                   p.rg, p.rbb, p.rm, p.rv,
                                                   R, M, cinP, coutT, coutS);
            res = R;
        }

        // tcn conv3x3 + bias + BN + residual + PReLU
        {
            int tot = coutP * 9 * cinP;
            k_pack_w3x3<<<(tot + 255) / 256, 256, 0, stream>>>(
                p.cw, WP3, coutT, coutP, cinT, cinP);
            dim3 g2(M / 128, coutP / 64);
            k_conv3x3_wmma<<<g2, 256, 0, stream>>>(H2, WP3, p.cb,
                                                   p.g, p.b, p.m, p.v,
                                                   res, p.pr, ZP, Y,
                                                   N, T, V, cinP, coutT, coutS);
        }

        // inter-layer temporal conv + BN + ReLU
        const int kt = (s < 3) ? 3 : T;       // final stage collapses time
        {
            int tot = coutP * kt * coutS;
            k_pack_wdown<<<(tot + 255) / 256, 256, 0, stream>>>(
                p.dw, WPD, coutT, coutP, coutT, coutS, kt);
        }
        if (s < 3) {
            const int Tout = T / 2;
            const int Md = N * Tout * V;
            dim3 g3(Md / 128, coutP / 64);
            k_down_wmma<<<g3, 256, 0, stream>>>(Y, WPD,
                                                p.dg, p.db, p.dm, p.dv, ZP,
                                                X, N, T, Tout, V, coutS,
                                                coutT, coutS, kt, 2, 1);
            T = Tout;
        } else {
            // output [N,1,V,2] == (N,V,C) flat, unpadded -> write d_out directly
            const int Md = N * 1 * V;
            dim3 g3(Md / 128, coutP / 64);
            k_down_wmma<<<g3, 256, 0, stream>>>(Y, WPD,
                                                p.dg, p.db, p.dm, p.dv, ZP,
                                                (float*)d_out, N, T, 1, V, coutS,
                                                coutT, coutT, kt, 1, 0);
        }
    }
}